// CLAHE_22574348108295
// MI455X (gfx1250) — compile-verified
//
#include <hip/hip_runtime.h>

// CLAHE for B=8, H=W=2048, 8x8 grid of 256x256 tiles, 256 bins.
// Kernel 1: per-tile histogram (8 per-wave LDS sub-hists) + clip/redistribute
//           + cumsum via V_WMMA_F32_16X16X4_F32 (two chained 16x16x16 matmuls)
//           -> fp32 LUT[512][256] in d_ws.
// Kernel 2: bilinear LUT blend; 9 neighbor LUTs staged in LDS via
//           GLOBAL_LOAD_ASYNC_TO_LDS_B32 (ASYNCcnt), 4 ds gathers/pixel.

#define TILE   256
#define NB     256
#define GY     8
#define GX     8
#define BATCH  8
#define Hv     2048
#define Wv     2048
#define CLIP_LIM 1024          // max(int(4.0 * 65536 / 256), 1)
#define AREA   (TILE * TILE)

typedef float v2f __attribute__((ext_vector_type(2)));
typedef float v8f __attribute__((ext_vector_type(8)));

#if __has_builtin(__builtin_amdgcn_global_load_async_to_lds_b32)
#define HAVE_ASYNC_LDS 1
typedef __attribute__((address_space(1))) int GInt;   // global (AS1)
typedef __attribute__((address_space(3))) int LInt;   // LDS    (AS3)
#else
#define HAVE_ASYNC_LDS 0
#endif

__device__ __forceinline__ void wait_async_all() {
#if __has_builtin(__builtin_amdgcn_s_wait_asynccnt)
    __builtin_amdgcn_s_wait_asynccnt(0);
#else
    asm volatile("s_wait_asynccnt 0" ::: "memory");
#endif
}

__global__ __launch_bounds__(256) void clahe_hist_lut_kernel(
    const float* __restrict__ img, float* __restrict__ lut)
{
    __shared__ unsigned hist8[8 * NB];   // per-wave sub-histograms
    __shared__ int      hbuf[NB];        // clipped histogram
    __shared__ int      ebuf[NB];        // per-bin clip excess
    __shared__ float    tbuf[16];        // row totals between the two matmuls

    const int tid = threadIdx.x;
    const int blk = blockIdx.x;          // 0..511 : tile id
    const int b   = blk >> 6;
    const int ty  = (blk >> 3) & 7;
    const int tx  = blk & 7;

    for (int i = tid; i < 8 * NB; i += 256) hist8[i] = 0u;
    __syncthreads();

    unsigned* myh = &hist8[(tid >> 5) * NB];     // one sub-hist per wave32
    const float* base = img + (size_t)b * Hv * Wv + (size_t)(ty * TILE) * Wv + tx * TILE;

    // 65536 px / tile; 256 thr * float4 = 1024 px per iter -> 64 iters
    #pragma unroll 4
    for (int it = 0; it < 64; ++it) {
        const int p = it * 1024 + tid * 4;
        const int r = p >> 8;
        const int c = p & 255;
        const float4 v = *(const float4*)(base + (size_t)r * Wv + c);
        atomicAdd(&myh[min(max((int)v.x, 0), 255)], 1u);
        atomicAdd(&myh[min(max((int)v.y, 0), 255)], 1u);
        atomicAdd(&myh[min(max((int)v.z, 0), 255)], 1u);
        atomicAdd(&myh[min(max((int)v.w, 0), 255)], 1u);
    }
    __syncthreads();

    // reduce the 8 sub-histograms; clip at CLIP_LIM
    {
        unsigned h = 0;
        #pragma unroll
        for (int w = 0; w < 8; ++w) h += hist8[w * NB + tid];
        int clipped = (int)h;
        int ex = 0;
        if (clipped > CLIP_LIM) { ex = clipped - CLIP_LIM; clipped = CLIP_LIM; }
        hbuf[tid] = clipped;
        ebuf[tid] = ex;
    }
    __syncthreads();

    // ---- wave0 only: redistribution + WMMA cumsum + LUT write ----
    if (tid < 32) {
        const int lane = tid;
        const int half = lane >> 4;      // which 16-lane half
        const int mn   = lane & 15;      // A-row m == B/D-col n for this lane

        // total excess (wave reduction)
        int ex = 0;
        #pragma unroll
        for (int j = 0; j < 8; ++j) ex += ebuf[lane * 8 + j];
        #pragma unroll
        for (int off = 16; off > 0; off >>= 1) ex += __shfl_xor(ex, off, 32);

        const int base_add = ex >> 8;        // excess // 256
        const int residual = ex & 255;       // excess %  256
        const int step     = (residual > 0) ? max(NB / residual, 1) : NB;

        // Matmul 1: P = S * U   (U[k][n] = k<=n)  -> intra-row inclusive prefix
        // S[m][k] = redistributed count of bin m*16+k, as exact f32.
        v8f acc = {};
        #pragma unroll
        for (int s = 0; s < 4; ++s) {
            const int k0 = 4 * s + 2 * half;     // K slice per A/B layout
            const int b0 = mn * 16 + k0;
            int v0 = hbuf[b0]     + base_add;
            int v1 = hbuf[b0 + 1] + base_add;
            if (residual > 0 && (b0 % step) == 0)       v0 += 1;
            if (residual > 0 && ((b0 + 1) % step) == 0) v1 += 1;
            v2f a, bm;
            a.x = (float)v0;
            a.y = (float)v1;
            bm.x = (k0     <= mn) ? 1.0f : 0.0f;
            bm.y = (k0 + 1 <= mn) ? 1.0f : 0.0f;
            acc = __builtin_amdgcn_wmma_f32_16x16x4_f32(
                false, a, false, bm, (short)0, acc, false, false);
        }

        // row totals t[r] = P[r][15]  (col-15 lives in lanes 15 / 31)
        if (mn == 15) {
            #pragma unroll
            for (int j = 0; j < 8; ++j) tbuf[j + 8 * half] = acc[j];
        }
        asm volatile("s_wait_dscnt 0" ::: "memory"); // same-wave LDS is in-order

        // Matmul 2: cdf = Lo * T + P   (Lo[m][k] = k<m, T[k][n] = t[k])
        #pragma unroll
        for (int s = 0; s < 4; ++s) {
            const int k0 = 4 * s + 2 * half;
            v2f a, bm;
            a.x = (k0     < mn) ? 1.0f : 0.0f;
            a.y = (k0 + 1 < mn) ? 1.0f : 0.0f;
            bm.x = tbuf[k0];
            bm.y = tbuf[k0 + 1];
            acc = __builtin_amdgcn_wmma_f32_16x16x4_f32(
                false, a, false, bm, (short)0, acc, false, false);
        }

        // LUT = clip(round(cdf * 255/area), 0, 255)
        const float scale = 255.0f / (float)AREA;
        float* lt = lut + (size_t)blk * NB;
        #pragma unroll
        for (int j = 0; j < 8; ++j) {
            const int row = j + 8 * half;
            const float lv = fminf(fmaxf(rintf(acc[j] * scale), 0.0f), 255.0f);
            lt[row * 16 + mn] = lv;
        }
    }
}

__global__ __launch_bounds__(256) void clahe_apply_kernel(
    const float* __restrict__ img, const float* __restrict__ lut,
    float* __restrict__ out)
{
    __shared__ float llut[9 * NB];       // 3x3 neighbor LUTs, 9 KiB

    const int tid   = threadIdx.x;
    const int id    = blockIdx.x;        // 0..4095
    const int tile  = id >> 3;
    const int chunk = id & 7;            // 32-row slab within the tile
    const int b  = tile >> 6;
    const int ty = (tile >> 3) & 7;
    const int tx = tile & 7;

    // stage 9 neighbor LUTs (clamped grid coords) into LDS.
    // Preferred path: direct global->LDS async copy (no VGPR round-trip).
    #pragma unroll
    for (int slot = 0; slot < 9; ++slot) {
        const int dy = slot / 3, dx = slot % 3;
        const int gy = min(max(ty - 1 + dy, 0), GY - 1);
        const int gx = min(max(tx - 1 + dx, 0), GX - 1);
        const float* g = &lut[(size_t)(((b * GY + gy) * GX) + gx) * NB + tid];
        float*       l = &llut[slot * NB + tid];
#if HAVE_ASYNC_LDS
        __builtin_amdgcn_global_load_async_to_lds_b32((GInt*)g, (LInt*)l, 0, 0);
#else
        *l = *g;
#endif
    }
#if HAVE_ASYNC_LDS
    wait_async_all();
#endif
    __syncthreads();

    const int c0   = (tid & 63) * 4;     // col within tile (float4 granularity)
    const int rsub = tid >> 6;           // 4 rows processed per pass
    const int x    = tx * TILE + c0;

    // per-column interpolation params (fixed per thread)
    float wx[4]; int sx1[4], sx2[4];
    #pragma unroll
    for (int j = 0; j < 4; ++j) {
        const float xf  = (float)(x + j) * (1.0f / TILE) - 0.5f;
        const float x1f = floorf(xf);
        wx[j] = xf - x1f;
        const int ix1 = min(max((int)x1f, 0), GX - 1);
        const int ix2 = min(max((int)x1f + 1, 0), GX - 1);
        sx1[j] = (ix1 - tx + 1) * NB;    // LDS slot-x offset
        sx2[j] = (ix2 - tx + 1) * NB;
    }

    const float* src = img + (size_t)b * Hv * Wv;
    float*       dst = out + (size_t)b * Hv * Wv;

    for (int rr = 0; rr < 8; ++rr) {
        const int y = ty * TILE + chunk * 32 + rr * 4 + rsub;
        const float yf  = (float)y * (1.0f / TILE) - 0.5f;
        const float y1f = floorf(yf);
        const float wy  = yf - y1f;
        const int iy1 = min(max((int)y1f, 0), GY - 1);
        const int iy2 = min(max((int)y1f + 1, 0), GY - 1);
        const int ry1 = (iy1 - ty + 1) * 3 * NB;
        const int ry2 = (iy2 - ty + 1) * 3 * NB;

        const float4 v = *(const float4*)(src + (size_t)y * Wv + x);
        const float pv[4] = {v.x, v.y, v.z, v.w};
        float res[4];
        #pragma unroll
        for (int j = 0; j < 4; ++j) {
            const int bin = min(max((int)pv[j], 0), 255);
            const float l11 = llut[ry1 + sx1[j] + bin];
            const float l12 = llut[ry1 + sx2[j] + bin];
            const float l21 = llut[ry2 + sx1[j] + bin];
            const float l22 = llut[ry2 + sx2[j] + bin];
            const float top = l11 + (l12 - l11) * wx[j];
            const float bot = l21 + (l22 - l21) * wx[j];
            const float r2  = top + (bot - top) * wy;
            res[j] = fminf(fmaxf(rintf(r2), 0.0f), 255.0f);
        }
        const float4 o = {res[0], res[1], res[2], res[3]};
        *(float4*)(dst + (size_t)y * Wv + x) = o;
    }
}

extern "C" void kernel_launch(void* const* d_in, const int* in_sizes, int n_in,
                              void* d_out, int out_size, void* d_ws, size_t ws_size,
                              hipStream_t stream)
{
    const float* img = (const float*)d_in[0];
    float*       lut = (float*)d_ws;         // 512 * 256 fp32 = 512 KiB
    float*       outp = (float*)d_out;

    clahe_hist_lut_kernel<<<BATCH * GY * GX, 256, 0, stream>>>(img, lut);
    clahe_apply_kernel<<<BATCH * GY * GX * 8, 256, 0, stream>>>(img, lut, outp);
}